// Pointnet2_16587163697981
// MI455X (gfx1250) — compile-verified
//
#include <hip/hip_runtime.h>

typedef __attribute__((ext_vector_type(2))) float v2f;
typedef __attribute__((ext_vector_type(8))) float v8f;

#define BN_EPS 1e-5f

// ---------------- transpose (B,3,N) -> (B,N,3) ----------------
__global__ __launch_bounds__(256)
void transpose_kernel(const float* __restrict__ xyz, float* __restrict__ pts, int B, int N) {
  size_t i = (size_t)blockIdx.x * blockDim.x + threadIdx.x;
  size_t tot = (size_t)B * N * 3;
  if (i >= tot) return;
  int c = (int)(i % 3);
  int n = (int)((i / 3) % N);
  int b = (int)(i / (3 * (size_t)N));
  pts[i] = xyz[((size_t)b * 3 + c) * N + n];
}

// ---------------- farthest point sampling: 1 block per batch ----------------
extern __shared__ float fps_smem[];
__global__ __launch_bounds__(256)
void fps_kernel(const float* __restrict__ pts, int* __restrict__ fps_idx,
                float* __restrict__ new_xyz, int N, int S) {
  int b = blockIdx.x;
  int tid = threadIdx.x;
  float* dist = fps_smem;                    // N floats
  float* rmax = fps_smem + N;                // 256 floats
  int*   rarg = (int*)(fps_smem + N + 256);  // 256 ints
  const float* P = pts + (size_t)b * N * 3;
  for (int i = tid; i < N; i += 256) dist[i] = 1e10f;
  __syncthreads();
  int far = 0;
  for (int it = 0; it < S; ++it) {
    float cx = P[(size_t)far * 3 + 0];
    float cy = P[(size_t)far * 3 + 1];
    float cz = P[(size_t)far * 3 + 2];
    if (tid == 0) {
      fps_idx[(size_t)b * S + it] = far;
      float* o = new_xyz + ((size_t)b * S + it) * 3;
      o[0] = cx; o[1] = cy; o[2] = cz;
    }
    float best = -1.0f; int bi = tid;
    for (int i = tid; i < N; i += 256) {
      float dx = P[(size_t)i * 3 + 0] - cx;
      float dy = P[(size_t)i * 3 + 1] - cy;
      float dz = P[(size_t)i * 3 + 2] - cz;
      float d = dx * dx + dy * dy + dz * dz;
      float nd = fminf(dist[i], d);
      dist[i] = nd;
      if (nd > best) { best = nd; bi = i; }
    }
    rmax[tid] = best; rarg[tid] = bi;
    __syncthreads();
    for (int s = 128; s > 0; s >>= 1) {
      if (tid < s) {
        float a = rmax[tid], bb = rmax[tid + s];
        if (bb > a || (bb == a && rarg[tid + s] < rarg[tid])) {
          rmax[tid] = bb; rarg[tid] = rarg[tid + s];
        }
      }
      __syncthreads();
    }
    far = rarg[0];
    __syncthreads();
  }
}

// ---------------- ball query: first Kn indices (ascending) within radius ----------------
__global__ __launch_bounds__(256)
void ballquery_kernel(const float* __restrict__ pts, const float* __restrict__ nxyz,
                      int* __restrict__ idx, float r2, int N, int S, int Kn, int total) {
  int q = blockIdx.x * blockDim.x + threadIdx.x;
  if (q >= total) return;
  int b = q / S;
  const float* P = pts + (size_t)b * N * 3;
  const float* c = nxyz + (size_t)q * 3;
  float cx = c[0], cy = c[1], cz = c[2];
  int* o = idx + (size_t)q * Kn;
  int cnt = 0, first = -1;
  for (int i = 0; i < N && cnt < Kn; ++i) {
    float dx = P[(size_t)i * 3 + 0] - cx;
    float dy = P[(size_t)i * 3 + 1] - cy;
    float dz = P[(size_t)i * 3 + 2] - cz;
    if (dx * dx + dy * dy + dz * dz <= r2) {
      if (first < 0) first = i;
      o[cnt++] = i;
    }
  }
  if (first < 0) first = 0;
  for (; cnt < Kn; ++cnt) o[cnt] = first;
}

// ---------------- gather + recenter (+ optional feature concat) ----------------
__global__ __launch_bounds__(256)
void gather_kernel(const float* __restrict__ pts, const float* __restrict__ feats,
                   const int* __restrict__ idx, const float* __restrict__ nxyz,
                   float* __restrict__ X, int N, int S, int Kn, int Cf, size_t rows) {
  size_t row = (size_t)blockIdx.x * blockDim.x + threadIdx.x;
  if (row >= rows) return;
  size_t q = row / Kn;
  int b = (int)(q / S);
  int id = idx[row];
  const float* p = pts + ((size_t)b * N + id) * 3;
  const float* c = nxyz + q * 3;
  int Cin = 3 + Cf;
  float* o = X + row * (size_t)Cin;
  o[0] = p[0] - c[0]; o[1] = p[1] - c[1]; o[2] = p[2] - c[2];
  for (int j = 0; j < Cf; ++j) o[3 + j] = feats[((size_t)b * N + id) * Cf + j];
}

// ---------------- group_all concat: rows of [xyz(3) | feat(Cf)] ----------------
__global__ __launch_bounds__(256)
void concat_kernel(const float* __restrict__ xyzp, const float* __restrict__ feat,
                   float* __restrict__ X, size_t rows, int Cf) {
  size_t r = (size_t)blockIdx.x * blockDim.x + threadIdx.x;
  if (r >= rows) return;
  float* o = X + r * (size_t)(3 + Cf);
  o[0] = xyzp[r * 3 + 0]; o[1] = xyzp[r * 3 + 1]; o[2] = xyzp[r * 3 + 2];
  for (int j = 0; j < Cf; ++j) o[3 + j] = feat[r * (size_t)Cf + j];
}

// ---------------- WMMA fp32 GEMM: C[M,N] = A[M,K] * W[N,K]^T + bias ----------------
// One wave computes a 16x64 output strip: 4 accumulators share one A fragment.
// No per-lane predication in the K loop (B addresses clamped, store masked), so
// EXEC is all-ones at every WMMA. Fragments are software-pipelined: loads for
// K-step k+4 are issued before the WMMAs for step k, so load latency overlaps
// matrix ops instead of stalling on s_wait_loadcnt 0 each step.
#define GEMM_NB 4
__global__ __launch_bounds__(32)
void gemm_wmma_kernel(const float* __restrict__ A, const float* __restrict__ W,
                      const float* __restrict__ bias, float* __restrict__ C,
                      int M, int Nn, int K) {
  const int lane = threadIdx.x;
  const int half = lane >> 4;
  const int l16 = lane & 15;
  const size_t row = (size_t)blockIdx.x * 16 + l16;   // M is a multiple of 16
  const float* Arow = A + row * (size_t)K;
  const int n0 = blockIdx.y * (16 * GEMM_NB);
  const float* Wrow[GEMM_NB];
#pragma unroll
  for (int t = 0; t < GEMM_NB; ++t) {
    int wn = n0 + t * 16 + l16;
    if (wn >= Nn) wn = Nn - 1;                        // clamp: load stays valid
    Wrow[t] = W + (size_t)wn * K;
  }
  v8f acc[GEMM_NB] = {};
  const int kb = half * 2;
  const int kend = K & ~3;                            // multiple-of-4 prefix
  if (kend > 0) {
    // prologue: fragments for k = 0
    v2f a_cur, b_cur[GEMM_NB];
    a_cur.x = Arow[kb];
    a_cur.y = Arow[kb + 1];
#pragma unroll
    for (int t = 0; t < GEMM_NB; ++t) {
      b_cur[t].x = Wrow[t][kb];
      b_cur[t].y = Wrow[t][kb + 1];
    }
    for (int k0 = 4; k0 < kend; k0 += 4) {
      int ka = k0 + kb;
      // issue next-step loads BEFORE consuming current fragments
      v2f a_nxt, b_nxt[GEMM_NB];
      a_nxt.x = Arow[ka];
      a_nxt.y = Arow[ka + 1];
#pragma unroll
      for (int t = 0; t < GEMM_NB; ++t) {
        b_nxt[t].x = Wrow[t][ka];
        b_nxt[t].y = Wrow[t][ka + 1];
      }
#pragma unroll
      for (int t = 0; t < GEMM_NB; ++t) {
        acc[t] = __builtin_amdgcn_wmma_f32_16x16x4_f32(false, a_cur, false, b_cur[t],
                                                       (short)0, acc[t], false, false);
      }
      a_cur = a_nxt;
#pragma unroll
      for (int t = 0; t < GEMM_NB; ++t) b_cur[t] = b_nxt[t];
    }
    // epilogue: last full K-step
#pragma unroll
    for (int t = 0; t < GEMM_NB; ++t) {
      acc[t] = __builtin_amdgcn_wmma_f32_16x16x4_f32(false, a_cur, false, b_cur[t],
                                                     (short)0, acc[t], false, false);
    }
  }
  if (kend < K) {                                     // uniform tail, EXEC untouched
    int ka = kend + kb;
    int i0 = (ka < K) ? ka : K - 1;
    int i1 = (ka + 1 < K) ? ka + 1 : K - 1;
    float m0 = (ka < K) ? 1.0f : 0.0f;
    float m1 = (ka + 1 < K) ? 1.0f : 0.0f;
    v2f a;
    a.x = Arow[i0] * m0;
    a.y = Arow[i1] * m1;
#pragma unroll
    for (int t = 0; t < GEMM_NB; ++t) {
      v2f bv;
      bv.x = Wrow[t][i0] * m0;
      bv.y = Wrow[t][i1] * m1;
      acc[t] = __builtin_amdgcn_wmma_f32_16x16x4_f32(false, a, false, bv,
                                                     (short)0, acc[t], false, false);
    }
  }
#pragma unroll
  for (int t = 0; t < GEMM_NB; ++t) {
    int col = n0 + t * 16 + l16;
    if (col < Nn) {
      float bb = bias ? bias[col] : 0.0f;
#pragma unroll
      for (int r = 0; r < 8; ++r) {
        size_t orow = (size_t)blockIdx.x * 16 + r + half * 8;
        C[orow * (size_t)Nn + col] = acc[t][r] + bb;
      }
    }
  }
}

// ---------------- BN statistics: one block per channel (deterministic) ----------------
__global__ __launch_bounds__(256)
void bn_stats_kernel(const float* __restrict__ X, float* __restrict__ stats, int M, int C) {
  int c = blockIdx.x;
  __shared__ float ss[256];
  __shared__ float sq[256];
  float s = 0.f, q2 = 0.f;
  for (int r = threadIdx.x; r < M; r += 256) {
    float v = X[(size_t)r * C + c];
    s += v; q2 += v * v;
  }
  ss[threadIdx.x] = s; sq[threadIdx.x] = q2;
  __syncthreads();
  for (int st = 128; st > 0; st >>= 1) {
    if ((int)threadIdx.x < st) {
      ss[threadIdx.x] += ss[threadIdx.x + st];
      sq[threadIdx.x] += sq[threadIdx.x + st];
    }
    __syncthreads();
  }
  if (threadIdx.x == 0) {
    float m = ss[0] / (float)M;
    stats[c] = m;
    stats[C + c] = sq[0] / (float)M - m * m;   // biased variance, like jnp.var
  }
}

__global__ __launch_bounds__(256)
void bn_apply_relu_kernel(float* __restrict__ X, const float* __restrict__ stats,
                          const float* __restrict__ g, const float* __restrict__ be,
                          size_t total, int C) {
  size_t i = (size_t)blockIdx.x * blockDim.x + threadIdx.x;
  if (i >= total) return;
  int c = (int)(i % C);
  float y = g[c] * (X[i] - stats[c]) * rsqrtf(stats[C + c] + BN_EPS) + be[c];
  X[i] = fmaxf(y, 0.0f);
}

// ---------------- max over sample dim: (Q,Kn,C) -> (Q,C) ----------------
__global__ __launch_bounds__(256)
void max_kernel(const float* __restrict__ X, float* __restrict__ Y,
                size_t Q, int Kn, int C) {
  size_t i = (size_t)blockIdx.x * blockDim.x + threadIdx.x;
  if (i >= Q * (size_t)C) return;
  size_t q = i / C;
  int c = (int)(i % C);
  float m = -3.402823466e38f;
  for (int k = 0; k < Kn; ++k) m = fmaxf(m, X[(q * Kn + k) * (size_t)C + c]);
  Y[i] = m;
}

static inline unsigned cdiv(size_t a, unsigned b) { return (unsigned)((a + b - 1) / b); }

extern "C" void kernel_launch(void* const* d_in, const int* in_sizes, int n_in,
                              void* d_out, int out_size, void* d_ws, size_t ws_size,
                              hipStream_t stream) {
  (void)in_sizes; (void)out_size; (void)ws_size;
  if (n_in < 47) return;
  const float* xyz = (const float*)d_in[0];
  // d_in layout: 0=xyz; sa1: (W,b,g,be)x3 at 1..12; sa2: 13..24; sa3: 25..36;
  // fc: W1,b1,g1,be1 (37..40), W2,b2,g2,be2 (41..44), Wt,bt (45,46)

  const int B = 16, N = 4096;
  const int S1 = 512, K1 = 32, S2 = 128, K2 = 64;
  const size_t M1 = (size_t)B * S1 * K1;   // 262144
  const size_t M2 = (size_t)B * S2 * K2;   // 131072
  const size_t M3 = (size_t)B * 128;       // 2048

  float* ws = (float*)d_ws;
  size_t off = 0;
  auto alloc = [&](size_t n) { size_t p = off; off += (n + 255) & ~(size_t)255; return p; };
  float* pts   = ws + alloc((size_t)B * N * 3);
  float* nx1   = ws + alloc((size_t)B * S1 * 3);
  int*   fidx1 = (int*)(ws + alloc((size_t)B * S1));
  float* nx2   = ws + alloc((size_t)B * S2 * 3);
  int*   fidx2 = (int*)(ws + alloc((size_t)B * S2));
  float* stats = ws + alloc(2048);
  float* l1p   = ws + alloc((size_t)B * S1 * 128);
  float* l2p   = ws + alloc((size_t)B * S2 * 256);
  float* X3    = ws + alloc(M3 * 259);
  float* h0    = ws + alloc(16 * 1024);
  float* h1    = ws + alloc(16 * 512);
  float* h2    = ws + alloc(16 * 256);
  int*   idxb  = (int*)(ws + alloc(M1));          // ball-query indices (reused)
  float* X0    = ws + alloc(M1 * 3);              // SA1 grouped input
  float* bufA  = ws + alloc(M1 * 128);            // ping
  float* bufB  = ws + alloc(M2 * 131 + 1024);     // pong (sized for SA2 input)

  auto gemm = [&](const float* A, int wi, float* C, size_t M, int Nn, int K) {
    dim3 grid((unsigned)(M / 16), (unsigned)((Nn + 16 * GEMM_NB - 1) / (16 * GEMM_NB)));
    gemm_wmma_kernel<<<grid, 32, 0, stream>>>(A, (const float*)d_in[wi],
                                              (const float*)d_in[wi + 1], C,
                                              (int)M, Nn, K);
  };
  auto bnrelu = [&](float* Xp, int wi, size_t M, int C) {
    bn_stats_kernel<<<(unsigned)C, 256, 0, stream>>>(Xp, stats, (int)M, C);
    size_t tot = M * (size_t)C;
    bn_apply_relu_kernel<<<cdiv(tot, 256), 256, 0, stream>>>(
        Xp, stats, (const float*)d_in[wi + 2], (const float*)d_in[wi + 3], tot, C);
  };

  // ---- transpose (B,3,N) -> (B,N,3) ----
  transpose_kernel<<<cdiv((size_t)B * N * 3, 256), 256, 0, stream>>>(xyz, pts, B, N);

  // ---- SA1 ----
  fps_kernel<<<B, 256, (N + 512) * sizeof(float), stream>>>(pts, fidx1, nx1, N, S1);
  ballquery_kernel<<<cdiv((size_t)B * S1, 256), 256, 0, stream>>>(
      pts, nx1, idxb, 0.2f * 0.2f, N, S1, K1, B * S1);
  gather_kernel<<<cdiv(M1, 256), 256, 0, stream>>>(pts, nullptr, idxb, nx1, X0,
                                                   N, S1, K1, 0, M1);
  gemm(X0, 1, bufA, M1, 64, 3);     bnrelu(bufA, 1, M1, 64);
  gemm(bufA, 5, bufB, M1, 64, 64);  bnrelu(bufB, 5, M1, 64);
  gemm(bufB, 9, bufA, M1, 128, 64); bnrelu(bufA, 9, M1, 128);
  max_kernel<<<cdiv((size_t)B * S1 * 128, 256), 256, 0, stream>>>(
      bufA, l1p, (size_t)B * S1, K1, 128);

  // ---- SA2 ----
  fps_kernel<<<B, 256, (S1 + 512) * sizeof(float), stream>>>(nx1, fidx2, nx2, S1, S2);
  ballquery_kernel<<<cdiv((size_t)B * S2, 256), 256, 0, stream>>>(
      nx1, nx2, idxb, 0.4f * 0.4f, S1, S2, K2, B * S2);
  gather_kernel<<<cdiv(M2, 256), 256, 0, stream>>>(nx1, l1p, idxb, nx2, bufB,
                                                   S1, S2, K2, 128, M2);
  gemm(bufB, 13, bufA, M2, 128, 131); bnrelu(bufA, 13, M2, 128);
  gemm(bufA, 17, bufB, M2, 128, 128); bnrelu(bufB, 17, M2, 128);
  gemm(bufB, 21, bufA, M2, 256, 128); bnrelu(bufA, 21, M2, 256);
  max_kernel<<<cdiv((size_t)B * S2 * 256, 256), 256, 0, stream>>>(
      bufA, l2p, (size_t)B * S2, K2, 256);

  // ---- SA3 (group_all) ----
  concat_kernel<<<cdiv(M3, 256), 256, 0, stream>>>(nx2, l2p, X3, M3, 256);
  gemm(X3, 25, bufA, M3, 256, 259);   bnrelu(bufA, 25, M3, 256);
  gemm(bufA, 29, bufB, M3, 512, 256); bnrelu(bufB, 29, M3, 512);
  gemm(bufB, 33, bufA, M3, 1024, 512); bnrelu(bufA, 33, M3, 1024);
  max_kernel<<<cdiv((size_t)B * 1024, 256), 256, 0, stream>>>(
      bufA, h0, (size_t)B, 128, 1024);

  // ---- FC head ----
  gemm(h0, 37, h1, 16, 512, 1024); bnrelu(h1, 37, 16, 512);
  gemm(h1, 41, h2, 16, 256, 512);  bnrelu(h2, 41, 16, 256);
  gemm(h2, 45, (float*)d_out, 16, 50, 256);
}